// OnlineTripletLoss_52630529245412
// MI455X (gfx1250) — compile-verified
//
#include <hip/hip_runtime.h>
#include <hip/hip_bf16.h>

typedef __attribute__((ext_vector_type(2))) float v2f;
typedef __attribute__((ext_vector_type(8))) float v8f;

#define BATCH   8192
#define DIM     512
#define TMARGIN 0.2f

#define KC   64             // K staged per LDS chunk
#define LDK  (KC + 4)       // stride 68 % 64 = 4 -> conflict-free frag reads
#define MB   64             // rows per workgroup (4 M-halves x 16)
#define NB   128            // cols per sweep iteration (2 N-groups x 64)

// ---------------------------------------------------------------- row norms
__global__ __launch_bounds__(256) void row_norm_kernel(const float* __restrict__ E,
                                                       float* __restrict__ sq) {
    const int lane = threadIdx.x & 31;
    const int wave = threadIdx.x >> 5;
    const int row  = blockIdx.x * 8 + wave;
    const float4* rp = (const float4*)(E + (size_t)row * DIM);
    float s = 0.f;
    #pragma unroll
    for (int i = lane; i < DIM / 4; i += 32) {
        float4 v = rp[i];
        s += v.x * v.x + v.y * v.y + v.z * v.z + v.w * v.w;
    }
    #pragma unroll
    for (int off = 16; off > 0; off >>= 1) s += __shfl_xor(s, off, 32);
    if (lane == 0) sq[row] = s;
}

// ------------------------------------------------- fused GEMM + row reduction
// PASS 1: hp[i] = max dist over {same label, d2>0};  hna[i] = min dist over negatives
// PASS 2: hs[i] = min dist over {negatives with dist > hp[i]}  (INF if none)
template <int PASS>
__global__ __launch_bounds__(256) void dist_reduce_kernel(
        const float* __restrict__ E, const int* __restrict__ labels,
        const float* __restrict__ sq, const float* __restrict__ hp_in,
        float* __restrict__ out_hp, float* __restrict__ out_hna,
        float* __restrict__ out_hs) {

    __shared__ float shA[MB][LDK];
    __shared__ float shB[NB][LDK];
    __shared__ float redA[2][MB];
    __shared__ float redB[2][MB];

    const float INF = __builtin_inff();
    const int tid  = threadIdx.x;
    const int lane = tid & 31;
    const int wave = tid >> 5;
    const int mq   = wave >> 1;        // 0..3 : which 16-row slice of the 64-row block
    const int ng   = wave & 1;         // 0..1 : which 64-col group of the 128-col tile
    const int half = lane >> 4;        // C/D layout: lanes 16-31 hold rows M+8
    const int l16  = lane & 15;
    const int mbase = blockIdx.x * MB;
    const int mrow0 = mbase + mq * 16;

    // per-lane row-side constants (accumulator VGPR k -> row mrow0 + k + 8*half)
    float sqm[8];  int labm[8];  float hpm[8];
    #pragma unroll
    for (int k = 0; k < 8; ++k) {
        const int gm = mrow0 + k + 8 * half;
        sqm[k]  = sq[gm];
        labm[k] = labels[gm];
        hpm[k]  = (PASS == 2) ? hp_in[gm] : 0.f;
    }

    // running per-(row,lane) reductions across the full column sweep
    float r0[8], r1[8];
    #pragma unroll
    for (int k = 0; k < 8; ++k) { r0[k] = (PASS == 1) ? -INF : INF; r1[k] = INF; }

    for (int nb = 0; nb < BATCH / NB; ++nb) {
        const int nbase = nb * NB;

        // this lane's 4 columns (one per accumulator j)
        int   ncol[4]; float sqn[4]; int labn[4];
        #pragma unroll
        for (int j = 0; j < 4; ++j) {
            ncol[j] = nbase + ng * 64 + j * 16 + l16;
            sqn[j]  = sq[ncol[j]];
            labn[j] = labels[ncol[j]];
        }

        v8f c0 = {}, c1 = {}, c2 = {}, c3 = {};
        for (int kc0 = 0; kc0 < DIM; kc0 += KC) {
            __syncthreads();
            // stage A block: MB x KC  (1024 float4, 4 per thread)
            for (int i = tid; i < MB * (KC / 4); i += 256) {
                const int r = i >> 4, c4 = i & 15;
                *(float4*)&shA[r][c4 * 4] =
                    *(const float4*)(E + (size_t)(mbase + r) * DIM + kc0 + c4 * 4);
            }
            // stage B block: NB x KC  (2048 float4, 8 per thread)
            for (int i = tid; i < NB * (KC / 4); i += 256) {
                const int r = i >> 4, c4 = i & 15;
                *(float4*)&shB[r][c4 * 4] =
                    *(const float4*)(E + (size_t)(nbase + r) * DIM + kc0 + c4 * 4);
            }
            __syncthreads();

            // A 16x4 f32 frag: VGPR0 = K 0/2, VGPR1 = K 1/3 (lanes 0-15 / 16-31)
            // one A frag feeds 4 independent accumulators -> 1.25 LDS b64 / WMMA
            #pragma unroll
            for (int kk = 0; kk < KC; kk += 4) {
                const int ko = kk + 2 * half;
                v2f a, b0, b1, b2, b3;
                a.x  = shA[mq * 16 + l16][ko];
                a.y  = shA[mq * 16 + l16][ko + 1];
                b0.x = shB[ng * 64 +  0 + l16][ko];  b0.y = shB[ng * 64 +  0 + l16][ko + 1];
                b1.x = shB[ng * 64 + 16 + l16][ko];  b1.y = shB[ng * 64 + 16 + l16][ko + 1];
                b2.x = shB[ng * 64 + 32 + l16][ko];  b2.y = shB[ng * 64 + 32 + l16][ko + 1];
                b3.x = shB[ng * 64 + 48 + l16][ko];  b3.y = shB[ng * 64 + 48 + l16][ko + 1];
                c0 = __builtin_amdgcn_wmma_f32_16x16x4_f32(false, a, false, b0, (short)0, c0, false, false);
                c1 = __builtin_amdgcn_wmma_f32_16x16x4_f32(false, a, false, b1, (short)0, c1, false, false);
                c2 = __builtin_amdgcn_wmma_f32_16x16x4_f32(false, a, false, b2, (short)0, c2, false, false);
                c3 = __builtin_amdgcn_wmma_f32_16x16x4_f32(false, a, false, b3, (short)0, c3, false, false);
            }
        }

        // tile epilogue: distances + online reductions over all 4 accumulators
        #pragma unroll
        for (int j = 0; j < 4; ++j) {
            const v8f c = (j == 0) ? c0 : (j == 1) ? c1 : (j == 2) ? c2 : c3;
            #pragma unroll
            for (int k = 0; k < 8; ++k) {
                const int gm = mrow0 + k + 8 * half;
                float d2 = sqm[k] + sqn[j] - 2.f * c[k];
                if (gm == ncol[j]) d2 = 0.f;                 // exact-zero diagonal
                const bool nz = d2 > 0.f;
                const float dist = nz ? sqrtf(d2) : 0.f;
                const bool same = (labm[k] == labn[j]);
                if (PASS == 1) {
                    if (same && nz) r0[k] = fmaxf(r0[k], dist);   // hardest positive
                    if (!same)      r1[k] = fminf(r1[k], dist);   // nearest negative
                } else {
                    if (!same && dist > hpm[k]) r0[k] = fminf(r0[k], dist);  // semi-hard
                }
            }
        }
    }

    // reduce across the 16 lanes sharing a row (xor<16 keeps halves apart)
    #pragma unroll
    for (int k = 0; k < 8; ++k) {
        #pragma unroll
        for (int off = 1; off < 16; off <<= 1) {
            if (PASS == 1) {
                r0[k] = fmaxf(r0[k], __shfl_xor(r0[k], off, 32));
                r1[k] = fminf(r1[k], __shfl_xor(r1[k], off, 32));
            } else {
                r0[k] = fminf(r0[k], __shfl_xor(r0[k], off, 32));
            }
        }
    }
    if (l16 == 0) {
        #pragma unroll
        for (int k = 0; k < 8; ++k) {
            const int rl = mq * 16 + k + 8 * half;
            redA[ng][rl] = r0[k];
            redB[ng][rl] = r1[k];
        }
    }
    __syncthreads();

    // combine the 2 column-group waves; this workgroup owns rows exclusively
    if (tid < MB) {
        if (PASS == 1) {
            out_hp[mbase + tid]  = fmaxf(redA[0][tid], redA[1][tid]);
            out_hna[mbase + tid] = fminf(redB[0][tid], redB[1][tid]);
        } else {
            out_hs[mbase + tid]  = fminf(redA[0][tid], redA[1][tid]);
        }
    }
}

// ---------------------------------------------------------------- final mean
__global__ __launch_bounds__(256) void finalize_kernel(
        const float* __restrict__ hp, const float* __restrict__ hna,
        const float* __restrict__ hs, float* __restrict__ out) {
    __shared__ float red[256];
    const float INF = __builtin_inff();
    float s = 0.f;
    for (int i = threadIdx.x; i < BATCH; i += 256) {
        const float h   = hp[i];
        const float hsv = hs[i];
        const float hn  = (hsv < INF) ? hsv : hna[i];
        s += fmaxf(h - hn + TMARGIN, 0.f);
    }
    red[threadIdx.x] = s;
    __syncthreads();
    for (int st = 128; st > 0; st >>= 1) {
        if (threadIdx.x < st) red[threadIdx.x] += red[threadIdx.x + st];
        __syncthreads();
    }
    if (threadIdx.x == 0) out[0] = red[0] / (float)BATCH;
}

extern "C" void kernel_launch(void* const* d_in, const int* in_sizes, int n_in,
                              void* d_out, int out_size, void* d_ws, size_t ws_size,
                              hipStream_t stream) {
    const float* E      = (const float*)d_in[0];
    const int*   labels = (const int*)d_in[1];

    float* sq  = (float*)d_ws;          // [BATCH]
    float* hp  = sq  + BATCH;           // hardest positive
    float* hna = hp  + BATCH;           // nearest negative (all)
    float* hs  = hna + BATCH;           // nearest semi-hard negative

    row_norm_kernel<<<BATCH / 8, 256, 0, stream>>>(E, sq);
    dist_reduce_kernel<1><<<BATCH / MB, 256, 0, stream>>>(E, labels, sq, nullptr,
                                                          hp, hna, hs);
    dist_reduce_kernel<2><<<BATCH / MB, 256, 0, stream>>>(E, labels, sq, hp,
                                                          hp, hna, hs);
    finalize_kernel<<<1, 256, 0, stream>>>(hp, hna, hs, (float*)d_out);
}